// RoiAlign_33578054320279
// MI455X (gfx1250) — compile-verified
//
#include <hip/hip_runtime.h>
#include <stdint.h>

#ifndef __has_builtin
#define __has_builtin(x) 0
#endif

#if __has_builtin(__builtin_amdgcn_tensor_load_to_lds) && __has_builtin(__builtin_amdgcn_s_wait_tensorcnt)
#define USE_TDM 1
#else
#define USE_TDM 0
#endif

#define GH 14
#define GW 14
#define NCELL (GH * GW)      // 196
#define C_TOT 256
#define H_TOT 128
#define W_TOT 128
#define HW_TOT (H_TOT * W_TOT)   // 16384
#define CHUNK 8              // channels per TDM tile
#define CH_PER_BLOCK 32      // channels per block
#define NCHUNK (CH_PER_BLOCK / CHUNK)
#define MAX_T 34             // max footprint extent (32px box -> <=34 incl. +1 neighbor)
#define TILE_ELTS (CHUNK * MAX_T * MAX_T)   // 9248 floats

typedef unsigned int u32x4 __attribute__((ext_vector_type(4)));
typedef int          i32x4 __attribute__((ext_vector_type(4)));
typedef int          i32x8 __attribute__((ext_vector_type(8)));

__global__ __launch_bounds__(256) void roi_align_tdm_kernel(
    const float* __restrict__ fm,
    const float* __restrict__ boxes,
    const int*   __restrict__ box_idx,
    float*       __restrict__ out)
{
    __shared__ __align__(16) float tile[2][TILE_ELTS];  // 73,984 B, double-buffered
    __shared__ float4 wq[NCELL];                        // packed bilinear weights
    __shared__ int4   oq[NCELL];                        // packed tile offsets

    const int m  = blockIdx.x;
    const int cb = blockIdx.y * CH_PER_BLOCK;

    const float bx1 = boxes[m * 4 + 0];
    const float by1 = boxes[m * 4 + 1];
    const float bx2 = boxes[m * 4 + 2];
    const float by2 = boxes[m * 4 + 3];
    const int   b   = box_idx[m];

    // Footprint of all bilinear taps for this box (clamped in-bounds -> no OOB needed).
    const int px0 = (int)floorf(bx1);
    const int py0 = (int)floorf(by1);
    const int pxe = min((int)floorf(bx2) + 1, W_TOT - 1);
    const int pye = min((int)floorf(by2) + 1, H_TOT - 1);
    const int pw  = pxe - px0 + 1;   // <= 34
    const int ph  = pye - py0 + 1;   // <= 34
    const int phpw = ph * pw;

    const float* src0 = fm + ((size_t)(b * C_TOT + cb) * HW_TOT
                              + (size_t)py0 * W_TOT + px0);

#if USE_TDM
    // Issues one 3D-tile DMA: X=pw (row), Y=ph (stride W), Z=CHUNK channels (stride H*W).
    auto tdm_issue = [&](int ck) {
        const uint64_t gaddr = (uint64_t)(uintptr_t)(src0 + (size_t)ck * CHUNK * HW_TOT);
        const uint32_t ldsa  = (uint32_t)(uintptr_t)(&tile[ck & 1][0]);

        u32x4 g0;
        g0[0] = 1u;                                   // count=1, user descriptor
        g0[1] = ldsa;                                 // lds_addr (bytes)
        g0[2] = (uint32_t)gaddr;                      // global_addr[31:0]
        g0[3] = (uint32_t)((gaddr >> 32) & 0x1FFFFFFu) | (2u << 30); // addr[56:32] | type=2

        i32x8 g1;
        g1[0] = (2 << 16);                            // data_size = 4 bytes
        g1[1] = (pw & 0xFFFF) << 16;                  // tensor_dim0[15:0]
        g1[2] = (ph & 0xFFFF) << 16;                  // dim0 hi=0 | tensor_dim1[15:0]
        g1[3] = (pw & 0xFFFF) << 16;                  // dim1 hi=0 | tile_dim0
        g1[4] = (ph & 0xFFFF) | (CHUNK << 16);        // tile_dim1 | tile_dim2
        g1[5] = W_TOT;                                // tensor_dim0_stride[31:0]
        g1[6] = (int)(((unsigned)HW_TOT & 0xFFFFu) << 16); // s0 hi=0 | tensor_dim1_stride[15:0]
        g1[7] = (int)((unsigned)HW_TOT >> 16);        // tensor_dim1_stride[47:16]

        i32x4 g2;
        g2[0] = CHUNK;                                // tensor_dim2
        g2[1] = 0;                                    // tensor_dim3 (unused)
        g2[2] = 0;                                    // tensor_dim2_stride (unused for 3D)
        g2[3] = 0;                                    // stride hi | tile_dim3=0

        i32x4 g3 = {0, 0, 0, 0};
        i32x8 g4 = {0, 0, 0, 0, 0, 0, 0, 0};          // VADDR4 slot: unused per ISA

        __builtin_amdgcn_tensor_load_to_lds(g0, g1, g2, g3, g4, 0);
    };

    if (threadIdx.x < 32) tdm_issue(0);               // overlap DMA(0) with meta compute
#endif

    // Per-cell bilinear metadata (computed once, reused across all 32 channels).
    if (threadIdx.x < NCELL) {
        const int cell = threadIdx.x;
        const int gy = cell / GW;
        const int gx = cell - gy * GW;
        const float sy = by1 + (float)gy * (1.0f / 13.0f) * (by2 - by1);
        const float sx = bx1 + (float)gx * (1.0f / 13.0f) * (bx2 - bx1);
        const float syf = floorf(sy);
        const float sxf = floorf(sx);
        const float ly = sy - syf;
        const float lx = sx - sxf;
        const float hy = 1.0f - ly;
        const float hx = 1.0f - lx;
        int y0 = min(max((int)syf, 0), H_TOT - 1);
        int x0 = min(max((int)sxf, 0), W_TOT - 1);
        int y1 = min(y0 + 1, H_TOT - 1);
        int x1 = min(x0 + 1, W_TOT - 1);
        const int y0l = y0 - py0, y1l = y1 - py0;
        const int x0l = x0 - px0, x1l = x1 - px0;
        wq[cell] = make_float4(hy * hx, hy * lx, ly * hx, ly * lx);
        oq[cell] = make_int4(y0l * pw + x0l, y0l * pw + x1l,
                             y1l * pw + x0l, y1l * pw + x1l);
    }

#if USE_TDM
    if (threadIdx.x < 32) __builtin_amdgcn_s_wait_tensorcnt(0);
#else
    __syncthreads();   // meta visible before fallback copy ordering below
    for (int i = threadIdx.x; i < CHUNK * phpw; i += 256) {
        const int z = i / phpw;
        const int r = i - z * phpw;
        const int y = r / pw;
        const int x = r - y * pw;
        tile[0][i] = src0[(size_t)z * HW_TOT + y * W_TOT + x];
    }
#endif
    __syncthreads();   // tile[0] + metadata published to all waves

    for (int ck = 0; ck < NCHUNK; ++ck) {
#if USE_TDM
        if (ck + 1 < NCHUNK && threadIdx.x < 32) tdm_issue(ck + 1);  // prefetch next tile
#endif
        // 8 channels * 196 cells = 1568 outputs; cell fastest -> coalesced 128B stores.
        const float* tbuf = &tile[ck & 1][0];
        float* op = out + ((size_t)(m * C_TOT + cb + ck * CHUNK)) * NCELL;
#pragma unroll
        for (int r = 0; r < (CHUNK * NCELL + 255) / 256; ++r) {
            const int i = threadIdx.x + r * 256;
            if (i < CHUNK * NCELL) {
                const int cl   = i / NCELL;
                const int cell = i - cl * NCELL;
                const float4 w4 = wq[cell];
                const int4   o4 = oq[cell];
                const float* t = tbuf + cl * phpw;
                float v = w4.x * t[o4.x];
                v = fmaf(w4.y, t[o4.y], v);
                v = fmaf(w4.z, t[o4.z], v);
                v = fmaf(w4.w, t[o4.w], v);
                op[cl * NCELL + cell] = v;
            }
        }

#if USE_TDM
        if (ck + 1 < NCHUNK) {
            if (threadIdx.x < 32) __builtin_amdgcn_s_wait_tensorcnt(0);
            __syncthreads();   // publish tile[(ck+1)&1]; also fences reads of tile[ck&1]
        }
#else
        __syncthreads();
        if (ck + 1 < NCHUNK) {
            const float* srcn = src0 + (size_t)(ck + 1) * CHUNK * HW_TOT;
            float* dst = &tile[(ck + 1) & 1][0];
            for (int i = threadIdx.x; i < CHUNK * phpw; i += 256) {
                const int z = i / phpw;
                const int r = i - z * phpw;
                const int y = r / pw;
                const int x = r - y * pw;
                dst[i] = srcn[(size_t)z * HW_TOT + y * W_TOT + x];
            }
            __syncthreads();
        }
#endif
    }
}

extern "C" void kernel_launch(void* const* d_in, const int* in_sizes, int n_in,
                              void* d_out, int out_size, void* d_ws, size_t ws_size,
                              hipStream_t stream) {
    const float* fm    = (const float*)d_in[0];
    const float* boxes = (const float*)d_in[1];
    const int*   bidx  = (const int*)d_in[2];
    float*       out   = (float*)d_out;

    const int M = in_sizes[1] / 4;                  // boxes is (M,4)
    dim3 grid(M, C_TOT / CH_PER_BLOCK, 1);          // (1024, 8)
    hipLaunchKernelGGL(roi_align_tdm_kernel, grid, dim3(256), 0, stream,
                       fm, boxes, bidx, out);
}